// SpringMass_81295140979550
// MI455X (gfx1250) — compile-verified
//
#include <hip/hip_runtime.h>
#include <stdint.h>

// ---------------- problem constants (from reference) ----------------
#define TSTEPS    4096          // scan length (T)
#define CHUNK     128           // steps per lane
#define NCHUNK    32            // chunks per row == lanes per wave (wave32)
#define ROWS_PB   4             // rows (waves) per block
#define THREADS   (ROWS_PB * 32)
#define CH_STRIDE 1040          // LDS bytes per chunk: 1024 payload + 16 pad (16B aligned)
#define ROW_LDS   (NCHUNK * CH_STRIDE)   // 33280 B per row
#define LDS_BYTES (ROWS_PB * ROW_LDS)    // 133120 B per block (<= 320KB/WGP, 2 blocks/WGP)

// RK4 closed-form coefficients.  g(s) = G s + h,  G = [[a, c],[1,0]],
// a = -C/M (const), c = -k/M (per step), h = (F/M, 0).
// Exact RK4 step:  s' = A s + b,  A = I + P G,  b = P h,
// P = dt I + dt^2/2 G + dt^3/6 G^2 + dt^4/24 G^3.
// Cayley-Hamilton (G^2 = aG + cI) gives  P = gamma*I + delta*G with
//   delta = DELTA0 + c*D4,   gamma = DT + c*G1,
// and A = alpha*I + beta*G,  alpha = 1 + delta*c,  beta = gamma + delta*a,
// b = (Fm*beta, Fm*delta),  Fm = F/M.
static constexpr double DTd = 0.01, Cd = 0.1, Md = 1.0;
static constexpr double Ad  = -Cd / Md;
static constexpr float  DT_F   = (float)DTd;
static constexpr float  A_F    = (float)Ad;
static constexpr float  INVM_F = (float)(1.0 / Md);
static constexpr float  D4_F   = (float)(DTd*DTd*DTd*DTd / 24.0);
static constexpr float  DELTA0 = (float)(DTd*DTd/2.0 + DTd*DTd*DTd*Ad/6.0
                                         + DTd*DTd*DTd*DTd*Ad*Ad/24.0);
static constexpr float  G1_F   = (float)(DTd*DTd*DTd/6.0 + DTd*DTd*DTd*DTd*Ad/24.0);

// Compute per-step closed-form coefficients from (k, F).
__device__ __forceinline__ void step_coeffs(float k, float F,
                                            float& c, float& alpha, float& beta,
                                            float& m00, float& m01,
                                            float& b0, float& b1, float& delta) {
  c = -k * INVM_F;
  delta = fmaf(c, D4_F, DELTA0);
  float gamma = fmaf(c, G1_F, DT_F);
  alpha = fmaf(delta, c, 1.0f);
  beta  = fmaf(delta, A_F, gamma);
  m00 = fmaf(beta, A_F, alpha);   // A = [[m00, m01],[beta, alpha]]
  m01 = beta * c;
  float Fm = F * INVM_F;
  b0 = Fm * beta;
  b1 = Fm * delta;
}

__launch_bounds__(THREADS, 2)
__global__ void spring_rk4_scan(const float* __restrict__ in,        // (B, T, 2)
                                const float* __restrict__ xdot_init, // (B, 1)
                                const float* __restrict__ x_init,    // (B, 1)
                                float* __restrict__ out)             // (B, T, 1)
{
  __shared__ __align__(16) unsigned char smem[LDS_BYTES];

  const int lane = threadIdx.x & 31;
  const int wave = threadIdx.x >> 5;
  const int row  = blockIdx.x * ROWS_PB + wave;

  unsigned char* wbase = smem + wave * ROW_LDS;
  const uint32_t wbase_lds = (uint32_t)(uintptr_t)wbase;   // low 32 bits == LDS offset

  // ---------------- Phase 0: async DMA this wave's row (32 KB) into LDS ----
  // 2048 x 16B segments per row; lane copies 64 of them, coalesced 512B/wave/iter.
  // LDS layout: chunk ch at ch*CH_STRIDE; 16B block s within chunk at
  // (s*16) ^ (((ch>>4)&1)<<4)  -> bank-conflict-free b64 reads later.
  {
    const uint64_t grow = (uint64_t)(uintptr_t)in + (uint64_t)row * (TSTEPS * 2u * 4u);
#pragma unroll 4
    for (int i = 0; i < 64; ++i) {
      const uint32_t seg   = (uint32_t)(i * 32 + lane);        // 0..2047
      const uint32_t chunk = seg >> 6;                          // 0..31
      const uint32_t sw    = ((chunk >> 4) & 1u) << 4;
      const uint32_t ldsa  = wbase_lds + chunk * CH_STRIDE + (((seg & 63u) << 4) ^ sw);
      const uint64_t ga    = grow + ((uint64_t)seg << 4);
      asm volatile("global_load_async_to_lds_b128 %0, %1, off"
                   :: "v"(ldsa), "v"(ga) : "memory");
    }
    asm volatile("s_wait_asynccnt 0" ::: "memory");
    // waves only read their own row's LDS region -> no block barrier needed
  }

  const uint32_t xsw = ((uint32_t)(lane >> 4) & 1u) << 4;      // this lane's swizzle
  unsigned char* cbase = wbase + lane * CH_STRIDE;

  // ---------------- Phase 1: per-lane chunk transform (affine 2x2 + vec2) ---
  float R00 = 1.f, R01 = 0.f, R10 = 0.f, R11 = 1.f, rb0 = 0.f, rb1 = 0.f;
#pragma unroll 4
  for (int j = 0; j < CHUNK; ++j) {
    const float2 kf = *(const float2*)(cbase + (((uint32_t)j << 3) ^ xsw));
    float c, alpha, beta, m00, m01, b0, b1, delta;
    step_coeffs(kf.x, kf.y, c, alpha, beta, m00, m01, b0, b1, delta);
    // running = step o running
    float n00 = fmaf(m00, R00, m01 * R10);
    float n01 = fmaf(m00, R01, m01 * R11);
    float n10 = fmaf(beta, R00, alpha * R10);
    float n11 = fmaf(beta, R01, alpha * R11);
    float nb0 = fmaf(m00, rb0, fmaf(m01, rb1, b0));
    float nb1 = fmaf(beta, rb0, fmaf(alpha, rb1, b1));
    R00 = n00; R01 = n01; R10 = n10; R11 = n11; rb0 = nb0; rb1 = nb1;
  }

  // ---------------- Phase 2: wave32 inclusive scan of affine transforms -----
#pragma unroll
  for (int off = 1; off < 32; off <<= 1) {
    float p00 = __shfl_up(R00, off), p01 = __shfl_up(R01, off);
    float p10 = __shfl_up(R10, off), p11 = __shfl_up(R11, off);
    float pb0 = __shfl_up(rb0, off), pb1 = __shfl_up(rb1, off);
    if (lane >= off) {   // cur o prev  (prev = earlier chunks)
      float n00 = fmaf(R00, p00, R01 * p10);
      float n01 = fmaf(R00, p01, R01 * p11);
      float n10 = fmaf(R10, p00, R11 * p10);
      float n11 = fmaf(R10, p01, R11 * p11);
      float nb0 = fmaf(R00, pb0, fmaf(R01, pb1, rb0));
      float nb1 = fmaf(R10, pb0, fmaf(R11, pb1, rb1));
      R00 = n00; R01 = n01; R10 = n10; R11 = n11; rb0 = nb0; rb1 = nb1;
    }
  }
  // exclusive prefix -> state entering this lane's chunk
  float e00 = __shfl_up(R00, 1), e01 = __shfl_up(R01, 1);
  float e10 = __shfl_up(R10, 1), e11 = __shfl_up(R11, 1);
  float eb0 = __shfl_up(rb0, 1), eb1 = __shfl_up(rb1, 1);
  if (lane == 0) { e00 = 1.f; e01 = 0.f; e10 = 0.f; e11 = 1.f; eb0 = 0.f; eb1 = 0.f; }
  const float xdi = xdot_init[row];
  const float xi  = x_init[row];
  float xd = fmaf(e00, xdi, fmaf(e01, xi, eb0));
  float x  = fmaf(e10, xdi, fmaf(e11, xi, eb1));

  // ---------------- Phase 3: replay chunk, write xddot into consumed slots --
#pragma unroll 4
  for (int j = 0; j < CHUNK; ++j) {
    unsigned char* addr = cbase + (((uint32_t)j << 3) ^ xsw);
    const float2 kf = *(const float2*)addr;
    float c, alpha, beta, m00, m01, b0, b1, delta;
    step_coeffs(kf.x, kf.y, c, alpha, beta, m00, m01, b0, b1, delta);
    float nxd = fmaf(m00, xd, fmaf(m01, x, b0));
    float nx  = fmaf(beta, xd, fmaf(alpha, x, b1));
    xd = nxd; x = nx;
    // xddot_t = (-C*xdot' - k*x' + F)/M  using *new* state (matches reference)
    float acc = fmaf(A_F, xd, fmaf(c, x, kf.y * INVM_F));
    *(float*)addr = acc;     // reuse k-slot of the just-consumed pair
  }

  // ---------------- Phase 4: coalesced drain LDS -> global (128B/instr) -----
  float* orow = out + (size_t)row * TSTEPS;
#pragma unroll 4
  for (int i = 0; i < TSTEPS / 32; ++i) {
    const int t = i * 32 + lane;
    const uint32_t ch = (uint32_t)t >> 7;
    const uint32_t j  = (uint32_t)t & 127u;
    const uint32_t sw = ((ch >> 4) & 1u) << 4;
    const float v = *(const float*)(wbase + ch * CH_STRIDE + ((j << 3) ^ sw));
    orow[t] = v;
  }
}

extern "C" void kernel_launch(void* const* d_in, const int* in_sizes, int n_in,
                              void* d_out, int out_size, void* d_ws, size_t ws_size,
                              hipStream_t stream) {
  (void)n_in; (void)out_size; (void)d_ws; (void)ws_size;
  const float* in        = (const float*)d_in[0];   // (B, T, 2) fp32
  const float* xdot_init = (const float*)d_in[1];   // (B, 1)
  const float* x_init    = (const float*)d_in[2];   // (B, 1)
  float*       out       = (float*)d_out;           // (B, T, 1)

  const int B = in_sizes[1];                        // 4096
  dim3 grid(B / ROWS_PB), block(THREADS);
  spring_rk4_scan<<<grid, block, 0, stream>>>(in, xdot_init, x_init, out);
}